// SignalGenerationLayer_39496519254378
// MI455X (gfx1250) — compile-verified
//
#include <hip/hip_runtime.h>

// ---------------------------------------------------------------------------
// qBOLD signal generation for MI455X (gfx1250, wave32).
// Quadrature contraction integral[v] = sum_j w_j * (1 - J0(c_v * tau * x_j))
// is mapped onto V_WMMA_F32_16X16X4_F32: A = per-voxel (1-J0) values
// (M=16 voxels, K=4 x-points), B = quadrature weights broadcast across all
// 16 columns, f32 accumulate.  Exploits evenness of J0 and of the blood term
// in tau: 11 taus -> 8 distinct |tau| quadratures (+ tau=0 trivial).
// ---------------------------------------------------------------------------

typedef __attribute__((ext_vector_type(2))) float v2f;
typedef __attribute__((ext_vector_type(8))) float v8f;

namespace {
constexpr double dGAMMA = 267500000.0;
constexpr double dB0    = 3.0;
constexpr double dDCHI  = 2.64e-07;
constexpr double dHCT   = 0.34;
constexpr double dTE    = 0.074;
constexpr double dR2T   = 11.5;
constexpr double dTR    = 3.0;
constexpr double dTI    = 1.21;
constexpr double dT1B   = 1.584;
constexpr double dNB    = 0.775;
constexpr double dTD    = 0.0045067;
constexpr double dPI    = 3.14159265358979323846;

// 1.5 * dw / oef  (arg prefactor)
constexpr float CDW15 = (float)(1.5 * (4.0 / 3.0) * dPI * dGAMMA * dB0 * dDCHI * dHCT);
constexpr float TER2T = (float)(dTE * dR2T);
constexpr float R2B0  = (float)(4.5 + 16.4 * dHCT);
constexpr float R2B2  = (float)(165.2 * dHCT + 55.7);
constexpr double dG0  = 4.0 / 45.0 * dHCT * (1.0 - dHCT) * (dDCHI * dB0) * (dDCHI * dB0);
constexpr float KBLOOD = (float)(0.5 * dGAMMA * dGAMMA * dG0 * dTD * dTD);
constexpr float TEOTD  = (float)(dTE / dTD);
constexpr float INVTD  = (float)(1.0 / dTD);
constexpr float TEf    = (float)dTE;
constexpr float NBf    = (float)dNB;
constexpr float DXf    = 1.0f / 32.0f;
constexpr float EXPA   = (float)(-(dTR - dTI) / dT1B);   // exp arg for m_bld
constexpr float EXPB   = (float)(-dTI / dT1B);
}

// NR / Abramowitz-Stegun J0, f32, branchless (matches reference formula).
__device__ __forceinline__ float j0_fast(float x) {
    float ax = fabsf(x);
    float y  = x * x;
    float p1 = fmaf(y, fmaf(y, fmaf(y, fmaf(y, fmaf(y, -184.9052456f, 77392.33017f),
                    -11214424.18f), 651619640.7f), -13362590354.0f), 57568490574.0f);
    float p2 = fmaf(y, fmaf(y, fmaf(y, fmaf(y, (y + 267.8532712f),
                    59272.64853f), 9494680.718f), 1029532985.0f), 57568490411.0f);
    float small_ = p1 * __builtin_amdgcn_rcpf(p2);

    float axb = fmaxf(ax, 8.0f);
    float inv = __builtin_amdgcn_rcpf(axb);
    float z   = 8.0f * inv;
    float y2  = z * z;
    float xx  = axb - 0.785398164f;
    float q1  = fmaf(y2, fmaf(y2, fmaf(y2, fmaf(y2, 2.093887211e-7f, -2.073370639e-6f),
                    2.734510407e-5f), -1.098628627e-3f), 1.0f);
    float q2  = fmaf(y2, fmaf(y2, fmaf(y2, fmaf(y2, -9.34935152e-8f, 7.621095161e-7f),
                    -6.911147651e-6f), 1.430488765e-4f), -1.562499995e-2f);
    float big = __builtin_amdgcn_sqrtf(0.636619772f * inv) *
                (__cosf(xx) * q1 - z * __sinf(xx) * q2);
    return (ax < 8.0f) ? small_ : big;
}

// quadrature weight * DX for x-point j (j = 0..31, j==0 -> 0).
// v_rcp_f32 instead of IEEE division: avoids div_scale/div_fixup sequences.
__device__ __forceinline__ float wdx(int j) {
    float x = (float)j * DXf;
    float w = (2.0f + x) * __builtin_amdgcn_sqrtf(1.0f - x) *
              __builtin_amdgcn_rcpf(x * x) * DXf;
    return (j == 0) ? 0.0f : w;
}

// select element sel (0..7) of a v8f with cndmasks (sel is per-lane)
__device__ __forceinline__ float pick8(v8f c, int sel) {
    float r = c[0];
    r = (sel == 1) ? c[1] : r;
    r = (sel == 2) ? c[2] : r;
    r = (sel == 3) ? c[3] : r;
    r = (sel == 4) ? c[4] : r;
    r = (sel == 5) ? c[5] : r;
    r = (sel == 6) ? c[6] : r;
    r = (sel == 7) ? c[7] : r;
    return r;
}

__global__ __launch_bounds__(256)
void qbold_signal_wmma(const float* __restrict__ inp,
                       const float* __restrict__ eps,
                       float* __restrict__ out, int nvox) {
    const int tid      = blockIdx.x * blockDim.x + threadIdx.x;
    const int lane     = threadIdx.x & 31;
    const int waveBase = (tid >> 5) << 4;            // 16 voxels per wave
    if (waveBase >= nvox) return;                    // wave-uniform exit

    const int koff = (lane >> 4) << 1;               // K offset: 0 (lanes 0-15) or 2
    const int mA   = lane & 15;                      // voxel this lane feeds into A
    const int mO   = (lane & 7) | ((lane & 16) >> 1);// voxel this lane "owns"
    const int vAi  = waveBase + mA;
    const int vOi  = waveBase + mO;
    const int vA   = (vAi < nvox) ? vAi : (nvox - 1);
    const int vO   = (vOi < nvox) ? vOi : (nvox - 1);

    const float oefA = inp[2 * vA + 0];
    const float oefO = inp[2 * vO + 0];
    const float dbvO = inp[2 * vO + 1];
    const float carg = CDW15 * oefA;                 // 1.5*dw for A-voxel

    // B operand: weights * DX (all 16 columns identical).  Lane holds K=koff,koff+1.
    v2f bmat[8];
#pragma unroll
    for (int ch = 0; ch < 8; ++ch) {
        bmat[ch].x = wdx(4 * ch + koff);
        bmat[ch].y = wdx(4 * ch + koff + 1);
    }

    // per-owned-voxel scalars
    const float r2bTE = fmaf(R2B2 * oefO, oefO, R2B0) * TEf;
    const float cbase = TEOTD + __builtin_amdgcn_sqrtf(0.25f + TEOTD) + 1.5f;
    const float m_bld = 1.0f - (2.0f - __expf(EXPA)) * __expf(EXPB);
    const float bwgt  = m_bld * NBf * dbvO;

    float ls[9];
    {   // |tau| = 0: integral == 0
        float C0     = cbase - 4.0f * __builtin_amdgcn_sqrtf(fmaf(TEf * TEf, INVTD, 0.25f));
        float tissue = __expf(-TER2T);
        float blood  = __expf(-r2bTE - KBLOOD * C0);
        ls[0] = __logf(fmaf(bwgt, blood - tissue, tissue));
    }

    const int sel = lane & 7;
#pragma unroll
    for (int k = 1; k <= 8; ++k) {                   // distinct |tau| = k*0.008
        const float tau = (float)(k * 0.008);
        const float t   = carg * tau;
        v8f acc = {};
#pragma unroll
        for (int ch = 0; ch < 8; ++ch) {
            float xb = (float)(4 * ch + koff) * DXf;
            v2f a;
            a.x = 1.0f - j0_fast(t * xb);
            a.y = 1.0f - j0_fast(t * (xb + DXf));
            // D(16 vox x 16) += A(16 vox x 4 xpts) * B(4 xpts x 16 weights)
            acc = __builtin_amdgcn_wmma_f32_16x16x4_f32(
                      false, a, false, bmat[ch], (short)0, acc, false, false);
        }
        float intg = pick8(acc, sel);                // integral for owned voxel

        float t1 = TEf + tau, t2 = TEf - tau;
        float Ck = cbase
                 - 2.0f * __builtin_amdgcn_sqrtf(fmaf(t1 * t1, INVTD, 0.25f))
                 - 2.0f * __builtin_amdgcn_sqrtf(fmaf(t2 * t2, INVTD, 0.25f));
        float tissue = __expf(fmaf(-dbvO * (1.0f / 3.0f), intg, -TER2T));
        float blood  = __expf(-r2bTE - KBLOOD * Ck);
        ls[k] = __logf(fmaf(bwgt, blood - tissue, tissue));
    }

    // Only owner lanes (0-7, 16-23) write out their voxel's 11 signals.
    if (((lane & 8) == 0) && (vOi < nvox)) {
        // taus index -> |tau|/0.008 : {-.016,-.008,0,.008,...,.064}
        const int kmap[11] = {2, 1, 0, 1, 2, 3, 4, 5, 6, 7, 8};
        const float sum  = ls[0] + 2.0f * (ls[1] + ls[2])
                         + ls[3] + ls[4] + ls[5] + ls[6] + ls[7] + ls[8];
        const float mean = sum * (1.0f / 11.0f) - ls[0];
        const float stdd = fabsf(mean) * (1.0f / 100.0f);   // / SNR
#pragma unroll
        for (int i = 0; i < 11; ++i) {
            float lsr = ls[kmap[i]] - ls[0];
            float nw  = sqrtf(2.0f * (float)(i + 1) / 11.0f);   // folded at compile time
            out[vO * 11 + i] = fmaf(eps[vO * 11 + i], stdd * nw, lsr);
        }
    }
}

extern "C" void kernel_launch(void* const* d_in, const int* in_sizes, int n_in,
                              void* d_out, int out_size, void* d_ws, size_t ws_size,
                              hipStream_t stream) {
    (void)n_in; (void)out_size; (void)d_ws; (void)ws_size;
    const float* inp = (const float*)d_in[0];   // (N_VOX, 2) f32
    const float* eps = (const float*)d_in[1];   // (N_VOX, 11) f32
    float* out = (float*)d_out;                 // (N_VOX, 11) f32
    const int nvox = in_sizes[0] / 2;
    const int voxPerBlock = 128;                // 8 waves * 16 voxels
    const int blocks = (nvox + voxPerBlock - 1) / voxPerBlock;
    qbold_signal_wmma<<<blocks, 256, 0, stream>>>(inp, eps, out, nvox);
}